// DummyTeModel_37632503447555
// MI455X (gfx1250) — compile-verified
//
#include <hip/hip_runtime.h>

typedef float v2f __attribute__((ext_vector_type(2)));
typedef float v8f __attribute__((ext_vector_type(8)));
typedef int   v4i __attribute__((ext_vector_type(4)));

#define BM 128
#define BN 128
#define BK 32
#define LDSTR 36  // BK+4: rows 16B-aligned, conflict-free fragment reads

#if defined(__gfx1250__) && __has_builtin(__builtin_amdgcn_global_load_async_to_lds_b128)
#define USE_ASYNC_LDS 1
#else
#define USE_ASYNC_LDS 0
#endif

__device__ __forceinline__ void wait_async0() {
#if __has_builtin(__builtin_amdgcn_s_wait_asynccnt)
    __builtin_amdgcn_s_wait_asynccnt(0);
#else
    asm volatile("s_wait_asynccnt 0" ::: "memory");
#endif
}

// Y[M,N] = X[M,K] @ W[N,K]^T + bias[N]
// Requires M%128==0, N%128==0, K%32==0 (true for every stage here).
__global__ __launch_bounds__(256) void gemm_bias_wmma_f32(
    const float* __restrict__ X, const float* __restrict__ W,
    const float* __restrict__ bias, float* __restrict__ Y,
    int M, int N, int K)
{
    const int tid  = threadIdx.x;
    const int lane = tid & 31;
    const int wave = tid >> 5;
    const int wm   = wave & 1;    // 2 wave-rows of 64
    const int wn   = wave >> 1;   // 4 wave-cols of 32
    const int half = lane >> 4;   // K-half (A/B) & M-half (C/D) selector
    const int ln16 = lane & 15;

    const int rowBase = blockIdx.y * BM;
    const int colBase = blockIdx.x * BN;

    v8f acc[4][2];
#pragma unroll
    for (int mi = 0; mi < 4; ++mi)
#pragma unroll
        for (int ni = 0; ni < 2; ++ni)
            acc[mi][ni] = (v8f){0.f, 0.f, 0.f, 0.f, 0.f, 0.f, 0.f, 0.f};

#if USE_ASYNC_LDS
    // ---- Async global->LDS path: no VGPR staging, 1 barrier per K-tile ----
    __shared__ float Xs[2][BM * LDSTR];
    __shared__ float Ws[2][BN * LDSTR];

    // Builtin signature (from clang diagnostics): (int4* global_src,
    // int4* lds_dst, imm offset, imm cpol) with cuda_device / cuda_shared
    // address spaces — pass plain generic pointers and let clang convert.
#define ISSUE(K0, B)                                                           \
    {                                                                          \
        _Pragma("unroll") for (int i = 0; i < 4; ++i)                          \
        {                                                                      \
            int u = tid + i * 256;                                             \
            int r = u >> 3;                                                    \
            int c = (u & 7) << 2;                                              \
            __builtin_amdgcn_global_load_async_to_lds_b128(                    \
                (v4i*)(X + (size_t)(rowBase + r) * K + (K0) + c),              \
                (v4i*)&Xs[B][r * LDSTR + c], 0, 0);                            \
            __builtin_amdgcn_global_load_async_to_lds_b128(                    \
                (v4i*)(W + (size_t)(colBase + r) * K + (K0) + c),              \
                (v4i*)&Ws[B][r * LDSTR + c], 0, 0);                            \
        }                                                                      \
    }

    ISSUE(0, 0);
    int buf = 0;
    for (int kt = 0; kt < K; kt += BK) {
        wait_async0();        // this wave's copies into `buf` are in LDS
        __syncthreads();      // all waves' copies done; all done reading buf^1

        const int knext = kt + BK;
        if (knext < K) ISSUE(knext, buf ^ 1);  // overlaps the WMMA burst below

#pragma unroll
        for (int kk = 0; kk < BK / 4; ++kk) {
            const int k = kk * 4 + 2 * half;
            v2f a[4], b[2];
#pragma unroll
            for (int s = 0; s < 4; ++s)
                a[s] = *(const v2f*)&Xs[buf][(wm * 64 + s * 16 + ln16) * LDSTR + k];
#pragma unroll
            for (int s = 0; s < 2; ++s)
                b[s] = *(const v2f*)&Ws[buf][(wn * 32 + s * 16 + ln16) * LDSTR + k];
#pragma unroll
            for (int mi = 0; mi < 4; ++mi)
#pragma unroll
                for (int ni = 0; ni < 2; ++ni)
                    acc[mi][ni] = __builtin_amdgcn_wmma_f32_16x16x4_f32(
                        false, a[mi], false, b[ni], (short)0, acc[mi][ni],
                        false, false);
        }
        buf ^= 1;
    }
#undef ISSUE

#else
    // ---- Fallback: register-staged global->LDS (proven path) ----
    __shared__ float Xs[BM * LDSTR];
    __shared__ float Ws[BN * LDSTR];
    float4 xr[4], wr[4];

#define GLOAD(K0)                                                              \
    {                                                                          \
        _Pragma("unroll") for (int i = 0; i < 4; ++i)                          \
        {                                                                      \
            int u = tid + i * 256;                                             \
            int r = u >> 3;                                                    \
            int c = (u & 7) << 2;                                              \
            xr[i] = *(const float4*)(X + (size_t)(rowBase + r) * K + (K0) + c);\
            wr[i] = *(const float4*)(W + (size_t)(colBase + r) * K + (K0) + c);\
        }                                                                      \
    }
#define STORE_LDS()                                                            \
    {                                                                          \
        _Pragma("unroll") for (int i = 0; i < 4; ++i)                          \
        {                                                                      \
            int u = tid + i * 256;                                             \
            int r = u >> 3;                                                    \
            int c = (u & 7) << 2;                                              \
            *(float4*)&Xs[r * LDSTR + c] = xr[i];                              \
            *(float4*)&Ws[r * LDSTR + c] = wr[i];                              \
        }                                                                      \
    }

    GLOAD(0);
    for (int kt = 0; kt < K; kt += BK) {
        STORE_LDS();
        __syncthreads();
        const int knext = kt + BK;
        if (knext < K) GLOAD(knext);
#pragma unroll
        for (int kk = 0; kk < BK / 4; ++kk) {
            const int k = kk * 4 + 2 * half;
            v2f a[4], b[2];
#pragma unroll
            for (int s = 0; s < 4; ++s)
                a[s] = *(const v2f*)&Xs[(wm * 64 + s * 16 + ln16) * LDSTR + k];
#pragma unroll
            for (int s = 0; s < 2; ++s)
                b[s] = *(const v2f*)&Ws[(wn * 32 + s * 16 + ln16) * LDSTR + k];
#pragma unroll
            for (int mi = 0; mi < 4; ++mi)
#pragma unroll
                for (int ni = 0; ni < 2; ++ni)
                    acc[mi][ni] = __builtin_amdgcn_wmma_f32_16x16x4_f32(
                        false, a[mi], false, b[ni], (short)0, acc[mi][ni],
                        false, false);
        }
        __syncthreads();
    }
#undef GLOAD
#undef STORE_LDS
#endif

    // Epilogue: C/D layout — lane n = ln16, row m = v + 8*half within each 16x16 tile
#pragma unroll
    for (int mi = 0; mi < 4; ++mi) {
#pragma unroll
        for (int ni = 0; ni < 2; ++ni) {
            const int gm = rowBase + wm * 64 + mi * 16 + 8 * half;
            const int gn = colBase + wn * 32 + ni * 16 + ln16;
            const float bv = bias[gn];
#pragma unroll
            for (int v = 0; v < 8; ++v)
                Y[(size_t)(gm + v) * N + gn] = acc[mi][ni][v] + bv;
        }
    }
}

static inline void run_gemm(const float* X, const float* W, const float* bias,
                            float* Y, int M, int N, int K, hipStream_t s)
{
    dim3 grid(N / BN, M / BM), blk(256);
    hipLaunchKernelGGL(gemm_bias_wmma_f32, grid, blk, 0, s, X, W, bias, Y, M, N, K);
}

extern "C" void kernel_launch(void* const* d_in, const int* in_sizes, int n_in,
                              void* d_out, int out_size, void* d_ws, size_t ws_size,
                              hipStream_t stream)
{
    (void)in_sizes; (void)n_in; (void)out_size; (void)ws_size;

    constexpr int H = 2048, I = 2 * H, G = 4;
    constexpr int T = 4 * 4096;        // B*S tokens
    constexpr int TG = T / G;          // tokens per group

    const float* x  = (const float*)d_in[0];   // [T, H]
    const float* w1 = (const float*)d_in[1];   // [I, H]
    const float* b1 = (const float*)d_in[2];   // [I]
    const float* w2 = (const float*)d_in[3];   // [H, I]
    const float* b2 = (const float*)d_in[4];   // [H]
    const float* gw[2][2] = { { (const float*)d_in[5], (const float*)d_in[7] },
                              { (const float*)d_in[9], (const float*)d_in[11] } };
    const float* gb[2][2] = { { (const float*)d_in[6], (const float*)d_in[8] },
                              { (const float*)d_in[10], (const float*)d_in[12] } };

    float* out = (float*)d_out;                // [T, H] — also reused as H-sized scratch
    float* wsI = (float*)d_ws;                 // [T, I] — I-sized scratch (256 MB)

    // Stage 1: h1 = x @ w1^T + b1            [T, I]
    run_gemm(x, w1, b1, wsI, T, I, H, stream);
    // Stage 2: h2 = h1 @ w2^T + b2           [T, H] (staged in out)
    run_gemm(wsI, w2, b2, out, T, H, I, stream);

    // Two grouped-MLP layers: out(H) -> wsI(I) -> out(H)
    for (int li = 0; li < 2; ++li) {
        for (int g = 0; g < G; ++g)  // g1: [TG,H] @ [I,H]^T -> [TG,I]
            run_gemm(out + (size_t)g * TG * H,
                     gw[li][0] + (size_t)g * I * H,
                     gb[li][0] + (size_t)g * I,
                     wsI + (size_t)g * TG * I, TG, I, H, stream);
        for (int g = 0; g < G; ++g)  // g2: [TG,I] @ [H,I]^T -> [TG,H]
            run_gemm(wsI + (size_t)g * TG * I,
                     gw[li][1] + (size_t)g * H * I,
                     gb[li][1] + (size_t)g * H,
                     out + (size_t)g * TG * H, TG, H, I, stream);
    }
}